// SinkhornLoss_16879221473401
// MI455X (gfx1250) — compile-verified
//
#include <hip/hip_runtime.h>

typedef __attribute__((ext_vector_type(2))) float v2f;
typedef __attribute__((ext_vector_type(8))) float v8f;

#define BB 4
#define NN 1024
#define DD 64
#define N_ITERS 14
#define BLUR2 0.0025f   // blur^p = 0.05^2

struct SoftminJob {
    const float* X;     // (B,NN,DD) rows side
    const float* Y;     // (B,NN,DD) reduce side
    const float* sqX;   // (B,NN) squared norms of X rows
    const float* sqY;   // (B,NN)
    const float* logw;  // (B,NN) log weights of reduce-side measure
    const float* pot;   // (B,NN) potential folded into h (always valid ptr)
    float        pscale;// 0.0f -> ignore pot (init pass), 1.0f otherwise
    float*       out;   // (B,NN)
};
struct JobTable { SoftminJob j[4]; };
struct PotTable { float* pot[4]; const float* tmp[4]; };

// ---------------------------------------------------------------------------
// Fused kernel: rebuild 16x16 tiles of C = max(0.5*(|x|^2+|y|^2) - x.y, 0)
// with V_WMMA_F32_16X16X4_F32 and reduce with an online logsumexp.
// One wave handles 16 output rows; B tiles are double-buffered in registers
// so the 16-WMMA chain + exp work hides the next tile's loads.
// Grid (NN/128, BB, 4 jobs), 256 thr = 8 waves. Control flow is uniform:
// EXEC is all-ones around every WMMA.
// ---------------------------------------------------------------------------
__global__ __launch_bounds__(256)
void softmin_pass(JobTable jt, const float* __restrict__ eps_list, int eps_idx) {
    const int b    = blockIdx.y;
    const SoftminJob jb = jt.j[blockIdx.z];
    const int lane = threadIdx.x & 31;
    const int wid  = threadIdx.x >> 5;
    const int n0   = (blockIdx.x * 8 + wid) * 16;

    const float eps     = eps_list[eps_idx];
    const float inv_eps = 1.0f / eps;

    const float* X    = jb.X    + (size_t)b * NN * DD;
    const float* Y    = jb.Y    + (size_t)b * NN * DD;
    const float* sqX  = jb.sqX  + b * NN;
    const float* sqY  = jb.sqY  + b * NN;
    const float* logw = jb.logw + b * NN;
    const float* pot  = jb.pot  + b * NN;
    const float  ps   = jb.pscale * inv_eps;   // h = logw + pot*ps

    const int lrow = lane & 15;     // tile column (B/C/D), or A row
    const int half = lane >> 4;     // K sub-slot selector
    const int koff = half * 2;      // A/B fragment K offset: lanes 16-31 hold K+2,K+3

    // Preload full A fragment set: 16 rows x K=64 -> 16 chunks of 2 VGPRs.
    v2f afrag[16];
    {
        const float* ap = X + (size_t)(n0 + lrow) * DD + koff;
        #pragma unroll
        for (int kc = 0; kc < 16; ++kc)
            afrag[kc] = *(const v2f*)(ap + kc * 4);
    }
    // 0.5*|x_row|^2 for the 8 rows this (vgpr,half) slot covers.
    float sqx8[8];
    #pragma unroll
    for (int v = 0; v < 8; ++v) sqx8[v] = 0.5f * sqX[n0 + v + 8 * half];

    float runmax[8], runsum[8];
    #pragma unroll
    for (int v = 0; v < 8; ++v) { runmax[v] = -3.0e38f; runsum[v] = 0.0f; }

    const float* brow = Y + (size_t)lrow * DD + koff;

    // One 16x16 column tile: 16 chained f32 WMMAs (2 accumulators) + online LSE.
    auto do_tile = [&](const v2f (&bf)[16], int m0) {
        v8f acc0 = {}, acc1 = {};
        #pragma unroll
        for (int kc = 0; kc < 16; kc += 2) {
            acc0 = __builtin_amdgcn_wmma_f32_16x16x4_f32(
                false, afrag[kc],     false, bf[kc],     (short)0, acc0, false, false);
            acc1 = __builtin_amdgcn_wmma_f32_16x16x4_f32(
                false, afrag[kc + 1], false, bf[kc + 1], (short)0, acc1, false, false);
        }
        const int m = m0 + lrow;                 // global column this lane owns
        const float hy = logw[m] + pot[m] * ps;  // branchless h[m]
        const float sy = 0.5f * sqY[m];
        #pragma unroll
        for (int v = 0; v < 8; ++v) {
            float c = fmaxf(sqx8[v] + sy - (acc0[v] + acc1[v]), 0.0f);
            float t = hy - c * inv_eps;          // h[m] - C[n,m]/eps
            float nm = fmaxf(runmax[v], t);
            runsum[v] = runsum[v] * __expf(runmax[v] - nm) + __expf(t - nm);
            runmax[v] = nm;
        }
    };

    // Double-buffered main loop over 64 column tiles (unrolled by 2).
    v2f bufA[16], bufB[16];
    #pragma unroll
    for (int kc = 0; kc < 16; ++kc) bufA[kc] = *(const v2f*)(brow + kc * 4);

    for (int m0 = 0; m0 < NN; m0 += 32) {
        const float* bpB = brow + (size_t)(m0 + 16) * DD;
        __builtin_prefetch(bpB + 32 * DD, 0, 3);      // global_prefetch_b8
        #pragma unroll
        for (int kc = 0; kc < 16; ++kc) bufB[kc] = *(const v2f*)(bpB + kc * 4);
        do_tile(bufA, m0);

        const int mA = (m0 + 32 < NN) ? (m0 + 32) : (NN - 16);  // clamp last (unused) load
        const float* bpA = brow + (size_t)mA * DD;
        #pragma unroll
        for (int kc = 0; kc < 16; ++kc) bufA[kc] = *(const v2f*)(bpA + kc * 4);
        do_tile(bufB, m0 + 16);
    }

    // Merge (max,sum) pairs across the 16 lanes of each half, then write rows.
    #pragma unroll
    for (int v = 0; v < 8; ++v) {
        float m1 = runmax[v], s1 = runsum[v];
        #pragma unroll
        for (int off = 1; off < 16; off <<= 1) {
            float m2 = __shfl_xor(m1, off, 32);
            float s2 = __shfl_xor(s1, off, 32);
            float nm = fmaxf(m1, m2);
            s1 = s1 * __expf(m1 - nm) + s2 * __expf(m2 - nm);
            m1 = nm;
        }
        if (lrow == v)
            jb.out[b * NN + n0 + v + 8 * half] = -eps * (m1 + __logf(s1));
    }
}

// --------------------------- small prep kernels ----------------------------
__global__ void prep_weights(const float* __restrict__ w1, const float* __restrict__ w2,
                             float* wn1, float* wn2, float* la, float* lb) {
    const int b = blockIdx.x, which = blockIdx.y;
    const float* w = which ? w2 : w1;
    float* wn = which ? wn2 : wn1;
    float* lw = which ? lb  : la;
    __shared__ float sd[256];
    float s = 0.0f;
    for (int i = threadIdx.x; i < NN; i += 256) s += w[b * NN + i];
    sd[threadIdx.x] = s; __syncthreads();
    for (int st = 128; st > 0; st >>= 1) {
        if (threadIdx.x < st) sd[threadIdx.x] += sd[threadIdx.x + st];
        __syncthreads();
    }
    float mass = sd[0];
    if (mass == 0.0f) mass = 1.0f;
    const float inv = 1.0f / mass;
    for (int i = threadIdx.x; i < NN; i += 256) {
        float v = w[b * NN + i] * inv;
        wn[b * NN + i] = v;
        lw[b * NN + i] = __logf(v);
    }
}

__global__ void row_sqnorm(const float* __restrict__ x, const float* __restrict__ y,
                           float* sqx, float* sqy) {
    const int which = blockIdx.y;
    const float* p = which ? y : x;
    float* o = which ? sqy : sqx;
    const int row = blockIdx.x * blockDim.x + threadIdx.x;
    if (row >= BB * NN) return;
    const float4* r = (const float4*)(p + (size_t)row * DD);
    float s = 0.0f;
    #pragma unroll
    for (int i = 0; i < DD / 4; ++i) {
        float4 v = r[i];
        s += v.x * v.x + v.y * v.y + v.z * v.z + v.w * v.w;
    }
    o[row] = s;
}

__global__ void compute_eps(const float* __restrict__ x, const float* __restrict__ y,
                            float* eps_list) {
    const int d = threadIdx.x;                   // 64 threads, one per dim
    float mx = -3.0e38f, mn = 3.0e38f;
    for (int p = 0; p < BB * NN; ++p) { float v = x[p * DD + d]; mx = fmaxf(mx, v); mn = fminf(mn, v); }
    for (int p = 0; p < BB * NN; ++p) { float v = y[p * DD + d]; mx = fmaxf(mx, v); mn = fminf(mn, v); }
    __shared__ float sd[DD];
    const float e = mx - mn;
    sd[d] = e * e; __syncthreads();
    if (d == 0) {
        float d2 = 0.0f;
        for (int i = 0; i < DD; ++i) d2 += sd[i];   // diameter^2
        float epsv = d2;                             // scaling^(p*i) = 0.25^i
        for (int i = 0; i < N_ITERS; ++i) { eps_list[i] = fmaxf(epsv, BLUR2); epsv *= 0.25f; }
        eps_list[N_ITERS] = BLUR2;                   // final extrapolation eps
    }
}

__global__ void update_pots(PotTable pt) {
    const int which = blockIdx.y;
    const int i = blockIdx.x * blockDim.x + threadIdx.x;
    if (i < BB * NN) {
        float* p = pt.pot[which];
        const float* t = pt.tmp[which];
        p[i] = 0.5f * (p[i] + t[i]);
    }
}

__global__ void final_loss(const float* __restrict__ wn1, const float* __restrict__ wn2,
                           const float* __restrict__ fba, const float* __restrict__ faa,
                           const float* __restrict__ gab, const float* __restrict__ gbb,
                           float* out) {
    __shared__ float sd[256];
    float s = 0.0f;
    for (int i = threadIdx.x; i < BB * NN; i += 256) {
        s += wn1[i] * (fba[i] - faa[i]);
        s += wn2[i] * (gab[i] - gbb[i]);
    }
    sd[threadIdx.x] = s; __syncthreads();
    for (int st = 128; st > 0; st >>= 1) {
        if (threadIdx.x < st) sd[threadIdx.x] += sd[threadIdx.x + st];
        __syncthreads();
    }
    if (threadIdx.x == 0) out[0] = sd[0] * (1.0f / BB);
}

// ---------------------------------------------------------------------------
extern "C" void kernel_launch(void* const* d_in, const int* in_sizes, int n_in,
                              void* d_out, int out_size, void* d_ws, size_t ws_size,
                              hipStream_t stream) {
    const float* x  = (const float*)d_in[0];
    const float* y  = (const float*)d_in[1];
    const float* w1 = (const float*)d_in[2];
    const float* w2 = (const float*)d_in[3];

    float* ws = (float*)d_ws;                    // ~230 KB total
    float* eps_list = ws;                        // 16
    float* la   = ws + 16;
    float* lb   = la   + BB * NN;
    float* wn1  = lb   + BB * NN;
    float* wn2  = wn1  + BB * NN;
    float* sqx  = wn2  + BB * NN;
    float* sqy  = sqx  + BB * NN;
    float* f_ba = sqy  + BB * NN;
    float* g_ab = f_ba + BB * NN;
    float* f_aa = g_ab + BB * NN;
    float* g_bb = f_aa + BB * NN;
    float* t_f  = g_bb + BB * NN;
    float* t_g  = t_f  + BB * NN;
    float* t_fa = t_g  + BB * NN;
    float* t_gb = t_fa + BB * NN;

    prep_weights<<<dim3(BB, 2), 256, 0, stream>>>(w1, w2, wn1, wn2, la, lb);
    row_sqnorm<<<dim3((BB * NN + 255) / 256, 2), 256, 0, stream>>>(x, y, sqx, sqy);
    compute_eps<<<1, DD, 0, stream>>>(x, y, eps_list);

    const dim3 smg(NN / 128, BB, 4);

    // init: potentials at eps_list[0], h = log-weights only (pscale = 0)
    JobTable jt0;
    jt0.j[0] = SoftminJob{x, y, sqx, sqy, lb, lb, 0.0f, f_ba};
    jt0.j[1] = SoftminJob{y, x, sqy, sqx, la, la, 0.0f, g_ab};
    jt0.j[2] = SoftminJob{x, x, sqx, sqx, la, la, 0.0f, f_aa};
    jt0.j[3] = SoftminJob{y, y, sqy, sqy, lb, lb, 0.0f, g_bb};
    softmin_pass<<<smg, 256, 0, stream>>>(jt0, eps_list, 0);

    for (int it = 0; it < N_ITERS; ++it) {
        JobTable st;
        st.j[0] = SoftminJob{x, y, sqx, sqy, lb, g_ab, 1.0f, t_f };
        st.j[1] = SoftminJob{y, x, sqy, sqx, la, f_ba, 1.0f, t_g };
        st.j[2] = SoftminJob{x, x, sqx, sqx, la, f_aa, 1.0f, t_fa};
        st.j[3] = SoftminJob{y, y, sqy, sqy, lb, g_bb, 1.0f, t_gb};
        softmin_pass<<<smg, 256, 0, stream>>>(st, eps_list, it);

        PotTable pt;
        pt.pot[0] = f_ba; pt.pot[1] = g_ab; pt.pot[2] = f_aa; pt.pot[3] = g_bb;
        pt.tmp[0] = t_f;  pt.tmp[1] = t_g;  pt.tmp[2] = t_fa; pt.tmp[3] = t_gb;
        update_pots<<<dim3((BB * NN + 255) / 256, 4), 256, 0, stream>>>(pt);
    }

    // final extrapolation at eps = blur^2 (eps_list[N_ITERS])
    JobTable jf;
    jf.j[0] = SoftminJob{x, y, sqx, sqy, lb, g_ab, 1.0f, t_f };
    jf.j[1] = SoftminJob{y, x, sqy, sqx, la, f_ba, 1.0f, t_g };
    jf.j[2] = SoftminJob{x, x, sqx, sqx, la, f_aa, 1.0f, t_fa};
    jf.j[3] = SoftminJob{y, y, sqy, sqy, lb, g_bb, 1.0f, t_gb};
    softmin_pass<<<smg, 256, 0, stream>>>(jf, eps_list, N_ITERS);

    final_loss<<<1, 256, 0, stream>>>(wn1, wn2, t_f, t_fa, t_g, t_gb, (float*)d_out);
}